// Attention_41979010351417
// MI455X (gfx1250) — compile-verified
//
#include <hip/hip_runtime.h>
#include <stdint.h>

// Attention: scores = enc·dec (per batch), softmax over L, context = attnᵀ·enc.
// L=1024, B=64, H=1024 fp32. Memory-bound: enc = 256MB. Fused single pass over
// enc (flash-style chunked softmax) -> ~11us at 23.3 TB/s instead of ~22us.

#define L_SEQ   1024
#define BATCH   64
#define HDIM    1024
#define CHUNK   16                 // encoder rows per block (one WMMA tile)
#define NCHUNK  (L_SEQ / CHUNK)    // 64 chunks per batch
#define ROWSTRIDE 4112             // 4096B row + 16B pad -> 4-bank skew per row
#define DEC_OFF  (CHUNK * ROWSTRIDE)        // 65792
#define SSH_OFF  (DEC_OFF + HDIM * 4)       // 69888
#define SMEM_BYTES (SSH_OFF + 64)           // 69952

typedef float    v2f  __attribute__((ext_vector_type(2)));
typedef float    v4f  __attribute__((ext_vector_type(4)));
typedef float    v8f  __attribute__((ext_vector_type(8)));
typedef uint32_t u32x4 __attribute__((ext_vector_type(4)));
typedef int32_t  i32x4 __attribute__((ext_vector_type(4)));
typedef int32_t  i32x8 __attribute__((ext_vector_type(8)));

#if __has_builtin(__builtin_amdgcn_tensor_load_to_lds)
#define HAVE_TDM 1
#else
#define HAVE_TDM 0
#endif

// Generic pointers into LDS carry the LDS byte offset in their low 32 bits
// (aperture in the high bits), which is what async/TDM LDS operands expect.
__device__ __forceinline__ uint32_t lds_lo32(const void* p) {
  return (uint32_t)(uintptr_t)p;
}

__device__ __forceinline__ void wait_async0() {
#if __has_builtin(__builtin_amdgcn_s_wait_asynccnt)
  __builtin_amdgcn_s_wait_asynccnt(0);
#else
  asm volatile("s_wait_asynccnt 0" ::: "memory");
#endif
}

#if HAVE_TDM
// Build a 2D tensor-DMA descriptor (ISA cdna5 §8) and issue TENSOR_LOAD_TO_LDS.
// data_size=4B elements; tile (tile_x × tile_y) from a tensor with row stride
// stride_x (elements); contiguous destination at LDS byte address lds_byte.
__device__ __forceinline__ void tdm_load_2d(uint32_t lds_byte, const void* gaddr,
                                            uint32_t tile_x, uint32_t tile_y,
                                            uint32_t tensor_x, uint32_t tensor_y,
                                            uint64_t stride_x) {
  uint64_t ga = (uint64_t)(uintptr_t)gaddr;
  u32x4 g0;
  g0[0] = 1u;                                                  // count=1 (valid), user mode
  g0[1] = lds_byte;                                            // lds_addr (bytes)
  g0[2] = (uint32_t)ga;                                        // global_addr[31:0]
  g0[3] = (uint32_t)((ga >> 32) & 0x01FFFFFFu) | 0x80000000u;  // addr[56:32] | type=2<<30
  i32x8 g1;
  g1[0] = (int)(2u << 16);                                     // wg_mask=0, data_size=2 (4B)
  g1[1] = (int)((tensor_x & 0xFFFFu) << 16);                   // tensor_dim0[15:0]
  g1[2] = (int)((tensor_x >> 16) | ((tensor_y & 0xFFFFu) << 16));
  g1[3] = (int)((tensor_y >> 16) | ((tile_x & 0xFFFFu) << 16));// tile_dim0
  g1[4] = (int)(tile_y & 0xFFFFu);                             // tile_dim1 (tile_dim2=0)
  g1[5] = (int)(uint32_t)(stride_x & 0xFFFFFFFFu);             // dim0_stride[31:0]
  g1[6] = (int)(uint32_t)((stride_x >> 32) & 0xFFFFu);         // dim0_stride[47:32]
  g1[7] = 0;
  i32x4 z4 = {0, 0, 0, 0};
#if __clang_major__ >= 23
  i32x8 z8 = {0, 0, 0, 0, 0, 0, 0, 0};
  __builtin_amdgcn_tensor_load_to_lds(g0, g1, z4, z4, z8, 0);
#else
  __builtin_amdgcn_tensor_load_to_lds(g0, g1, z4, z4, 0);
#endif
}
#endif

// One wave (32 lanes) per block; block = (chunk of 16 l-rows, one batch b).
// 1) async-stage 16 enc rows (64KB, bank-skewed) + dec row (TDM) into LDS
// 2) scores[16] tile via v_wmma_f32_16x16x4_f32, dec broadcast across N
// 3) chunk-local softmax partials: m_c, d_c, pc[h] = sum_l e^(s_l-m_c)*row_l[h]
__global__ void attn_scores_partial(const float* __restrict__ enc,
                                    const float* __restrict__ dec,
                                    float* __restrict__ ws_md,
                                    float* __restrict__ ws_pc) {
  extern __shared__ char smem[];
  const int lane  = threadIdx.x;       // 0..31
  const int chunk = blockIdx.x;        // 0..63
  const int b     = blockIdx.y;        // 0..63
  const int l0    = chunk * CHUNK;
  const uint32_t smem_base = lds_lo32(smem);

  // ---- stage 16 encoder rows via async global->LDS copies (ASYNCcnt) ----
  // Each lane moves 16B per op; 8 ops cover one 4KB row. Offset applies to
  // both the global and LDS address (rows contiguous within LDS row).
#pragma unroll
  for (int r = 0; r < CHUNK; ++r) {
    const float* g  = enc + ((size_t)(l0 + r) * BATCH + b) * HDIM + lane * 4;
    uint32_t ldst = smem_base + (uint32_t)(r * ROWSTRIDE) + (uint32_t)(lane * 16);
#pragma unroll
    for (int o = 0; o < 8; ++o) {
      asm volatile("global_load_async_to_lds_b128 %0, %1, off offset:%2"
                   :: "v"(ldst), "v"(g), "i"(o * 512) : "memory");
    }
  }

  // ---- stage decoder vector (4KB): Tensor Data Mover if available ----
#if HAVE_TDM
  tdm_load_2d(smem_base + DEC_OFF, dec + (size_t)b * HDIM,
              /*tile*/ HDIM, 1, /*tensor*/ HDIM, 1, /*stride*/ HDIM);
  __builtin_amdgcn_s_wait_tensorcnt(0);
#else
  {
    const float* g = dec + (size_t)b * HDIM + lane * 4;
    uint32_t ldst = smem_base + DEC_OFF + (uint32_t)(lane * 16);
#pragma unroll
    for (int o = 0; o < 8; ++o) {
      asm volatile("global_load_async_to_lds_b128 %0, %1, off offset:%2"
                   :: "v"(ldst), "v"(g), "i"(o * 512) : "memory");
    }
  }
#endif
  wait_async0();
  __syncthreads();

  // ---- WMMA scores: D[m][n] = sum_k A[m][k]*B[k][n], B columns = dec chunk ----
  // A 16x4 f32 layout: lane m=lane&15, K = vgpr + 2*(lane>>4) -> 8B at
  // row*(ROWSTRIDE) + k0*4 + 8*half. B mirrors with the dec vector.
  const int  half = lane >> 4;
  const char* ap  = smem + (lane & 15) * ROWSTRIDE + 8 * half;
  const char* bp  = smem + DEC_OFF + 8 * half;
  v8f acc = {0.f, 0.f, 0.f, 0.f, 0.f, 0.f, 0.f, 0.f};
#pragma unroll 8
  for (int k0 = 0; k0 < HDIM; k0 += 4) {
    v2f av = *(const v2f*)(ap + (size_t)k0 * 4);
    v2f bv = *(const v2f*)(bp + (size_t)k0 * 4);
    acc = __builtin_amdgcn_wmma_f32_16x16x4_f32(false, av, false, bv,
                                                (short)0, acc, false, false);
  }

  // ---- broadcast the 16 scores (lane0: M=0..7 in vgpr r, lane16: M=8..15) ----
  float* s_sh = (float*)(smem + SSH_OFF);
  if (lane == 0) {
#pragma unroll
    for (int r = 0; r < 8; ++r) s_sh[r] = acc[r];
  }
  if (lane == 16) {
#pragma unroll
    for (int r = 0; r < 8; ++r) s_sh[8 + r] = acc[r];
  }
  __syncthreads();

  // ---- chunk-local softmax partials (computed redundantly in all lanes) ----
  float m_c = s_sh[0];
#pragma unroll
  for (int j = 1; j < CHUNK; ++j) m_c = fmaxf(m_c, s_sh[j]);
  float w[CHUNK];
  float d_c = 0.f;
#pragma unroll
  for (int j = 0; j < CHUNK; ++j) { w[j] = __expf(s_sh[j] - m_c); d_c += w[j]; }

  // ---- partial context from LDS-resident rows (reads are bank-conflict-free) --
  v4f pca[8];
#pragma unroll
  for (int j = 0; j < 8; ++j) pca[j] = (v4f){0.f, 0.f, 0.f, 0.f};
#pragma unroll
  for (int l = 0; l < CHUNK; ++l) {
    const char* rp = smem + l * ROWSTRIDE;
    float wl = w[l];
#pragma unroll
    for (int j = 0; j < 8; ++j) {
      v4f v = *(const v4f*)(rp + (size_t)(lane * 16 + j * 512));
      pca[j] += wl * v;
    }
  }
  float* pc = ws_pc + (size_t)(b * NCHUNK + chunk) * HDIM;
#pragma unroll
  for (int j = 0; j < 8; ++j)
    *(v4f*)(pc + lane * 4 + j * 128) = pca[j];
  if (lane == 0) {
    ws_md[(b * NCHUNK + chunk) * 2 + 0] = m_c;
    ws_md[(b * NCHUNK + chunk) * 2 + 1] = d_c;
  }
}

// Merge chunk partials: M = max_c m_c, D = sum_c d_c*e^(m_c-M),
// context[b,h] = (1/D) * sum_c e^(m_c-M) * pc[c][h].
__global__ void attn_combine(const float* __restrict__ ws_md,
                             const float* __restrict__ ws_pc,
                             float* __restrict__ out) {
  const int b = blockIdx.x;
  const int t = threadIdx.x;  // 0..255
  __shared__ float e_sh[NCHUNK];
  __shared__ float inv_d;

  const float* md = ws_md + (size_t)b * NCHUNK * 2;
  float M = md[0];
  for (int c = 1; c < NCHUNK; ++c) M = fmaxf(M, md[2 * c]);
  if (t < NCHUNK) e_sh[t] = __expf(md[2 * t] - M);
  __syncthreads();
  if (t == 0) {
    float D = 0.f;
    for (int c = 0; c < NCHUNK; ++c) D += md[2 * c + 1] * e_sh[c];
    inv_d = 1.0f / D;
  }
  __syncthreads();
#pragma unroll
  for (int j = 0; j < HDIM / 256; ++j) {
    int h = t + j * 256;
    float acc = 0.f;
    for (int c = 0; c < NCHUNK; ++c)
      acc += e_sh[c] * ws_pc[(size_t)(b * NCHUNK + c) * HDIM + h];
    out[(size_t)b * HDIM + h] = acc * inv_d;
  }
}

extern "C" void kernel_launch(void* const* d_in, const int* in_sizes, int n_in,
                              void* d_out, int out_size, void* d_ws, size_t ws_size,
                              hipStream_t stream) {
  (void)in_sizes; (void)n_in; (void)out_size; (void)ws_size;
  const float* enc = (const float*)d_in[0];   // [L,B,H] fp32
  const float* dec = (const float*)d_in[1];   // [1,B,H] fp32
  float* out = (float*)d_out;                 // [B,H] fp32

  float* ws_md = (float*)d_ws;                       // 64*64*2 floats = 32 KB
  float* ws_pc = (float*)((char*)d_ws + 32768);      // 64*64*1024 floats = 16 MB

  dim3 grid1(NCHUNK, BATCH);
  attn_scores_partial<<<grid1, 32, SMEM_BYTES, stream>>>(enc, dec, ws_md, ws_pc);
  attn_combine<<<BATCH, 256, 0, stream>>>(ws_md, ws_pc, out);
}